// RobertaCrfForTokenClassification_23742579212748
// MI455X (gfx1250) — compile-verified
//
#include <hip/hip_runtime.h>
#include <hip/hip_bf16.h>
#include <math.h>

// Problem dims
#define BB 64
#define SS 512
#define HH 1024
#define LL 9
#define LP 16          // L padded to 16 (WMMA N dim)
#define BK 64          // K-tile over H staged in LDS
#define WROWS 10       // 9 real W rows + 1 zero row for lanes n >= 9
#define WSTRIDE 1028   // W LDS row stride (floats): banks 4n, conflict-free
#define HSTRIDE 68     // hidden LDS row stride (floats): 16B-aligned rows, banks 4r+k

typedef __attribute__((ext_vector_type(2))) float v2f;
typedef __attribute__((ext_vector_type(8))) float v8f;

// ---------------------------------------------------------------------------
// Kernel 1: emissions[b,s,l] = hidden[b,s,:] . W[l,:] + bias[l], stored with
// row stride LP=16. 4 waves/block, each wave owns a 16-row x 16-col tile and
// runs V_WMMA_F32_16X16X4_F32 over H in K=4 chunks (BK=64 staged in LDS).
// Lanes with n>=9 read a pre-zeroed LDS row, so the B fragment needs no
// per-iteration masking and the WMMA chain has no extra VALU dependency.
// ---------------------------------------------------------------------------
__global__ __launch_bounds__(128) void emis_gemm_wmma(
    const float* __restrict__ hidden, const float* __restrict__ W,
    const float* __restrict__ bias, float* __restrict__ ews) {
  __shared__ float Wl[WROWS * WSTRIDE];     // 10*1028*4  = 41,120 B
  __shared__ float Hl[4][16 * HSTRIDE];     // 4*16*68*4  = 17,408 B

  // Stage W (9 x 1024) into LDS with padded stride + one zero row, coalesced.
  for (int idx = threadIdx.x; idx < WROWS * HH; idx += 128) {
    int r = idx >> 10;          // /1024
    int c = idx & (HH - 1);
    Wl[r * WSTRIDE + c] = (r < LL) ? W[r * HH + c] : 0.0f;
  }
  __syncthreads();

  const int wave = threadIdx.x >> 5;
  const int lane = threadIdx.x & 31;
  const int l16  = lane & 15;
  const int khalf = (lane < 16) ? 0 : 2;      // ISA 16x4 fp32 A layout
  const int row0 = (blockIdx.x * 4 + wave) * 16;
  const float* hrow = hidden + (size_t)row0 * HH;
  float* lds = Hl[wave];

  // Lanes n>=9 fetch the zero row (same-address broadcast, conflict-free).
  const int nrow = (l16 < LL) ? l16 : LL;

  v8f acc = {};
  for (int t = 0; t < HH / BK; ++t) {
    const int h0 = t * BK;

    // Prefetch a slice of the next K-tile (emits global_prefetch_b8).
    if (t + 1 < HH / BK)
      __builtin_prefetch(hrow + (size_t)l16 * HH + h0 + BK + ((lane >> 4) * 32), 0, 3);

    __syncthreads();  // WAR: previous iteration's fragment reads done
    // Coalesced float4 stage of the 16 x 64 hidden tile into LDS.
    #pragma unroll
    for (int j = 0; j < 8; ++j) {
      int idx = j * 32 + lane;          // float4 units within tile
      int r = idx >> 4;                 // row 0..15
      int c = (idx & 15) << 2;          // col 0..60 step 4
      float4 v = *reinterpret_cast<const float4*>(hrow + (size_t)r * HH + h0 + c);
      *reinterpret_cast<float4*>(&lds[r * HSTRIDE + c]) = v;
    }
    __syncthreads();  // RAW: tile visible to all lanes

    #pragma unroll
    for (int kk = 0; kk < BK; kk += 4) {
      // A fragment: lane holds A[m=l16][kb..kb+1], kb = khalf (ISA layout)
      v2f a = *reinterpret_cast<const v2f*>(&lds[l16 * HSTRIDE + kk + khalf]);
      // B fragment: B[k][n] = W[n][h0+kk+k]; n>=9 reads the zero row
      v2f bm = *reinterpret_cast<const v2f*>(&Wl[nrow * WSTRIDE + h0 + kk + khalf]);
      acc = __builtin_amdgcn_wmma_f32_16x16x4_f32(
          /*neg_a=*/false, a, /*neg_b=*/false, bm,
          /*c_mod=*/(short)0, acc, /*reuse_a=*/false, /*reuse_b=*/false);
    }
  }

  // Epilogue: D layout — VGPR j: lanes 0-15 row j, lanes 16-31 row 8+j, col=lane%16
  const float bn = (l16 < LL) ? bias[l16] : 0.0f;
  const int rbase = row0 + ((lane < 16) ? 0 : 8);
  #pragma unroll
  for (int j = 0; j < 8; ++j)
    ews[(size_t)(rbase + j) * LP + l16] = acc[j] + bn;
}

// ---------------------------------------------------------------------------
// Kernel 2: per-batch CRF forward scan + gold score. One wave32 per batch.
// Lane (j,g): j = lane/3 (tag), g = lane%3 covers predecessors {3g..3g+2}.
// ---------------------------------------------------------------------------
__global__ __launch_bounds__(32) void crf_scan(
    const float* __restrict__ ews, const float* __restrict__ start_t,
    const float* __restrict__ end_t, const float* __restrict__ trans,
    const int* __restrict__ labels, const int* __restrict__ mask,
    float* __restrict__ llh) {
  const int b = blockIdx.x;
  const int lane = threadIdx.x;
  int jj = lane / 3; if (jj > 8) jj = 8;      // lanes 27-31 mirror lane 24
  const int g = (lane < 27) ? (lane % 3) : 0;

  // Transition columns this lane reduces: T[3g+q][jj]
  const float t0 = trans[(3 * g + 0) * LL + jj];
  const float t1 = trans[(3 * g + 1) * LL + jj];
  const float t2 = trans[(3 * g + 2) * LL + jj];

  const float* eb = ews + (size_t)b * SS * LP;
  const int* mb = mask + b * SS;
  const int* lb = labels + b * SS;

  float alpha = start_t[jj] + eb[jj];         // alpha0
  for (int t = 1; t < SS; ++t) {
    const float ej = eb[t * LP + jj];
    const int mt = mb[t];
    // gather alpha[i] for i in {3g, 3g+1, 3g+2}; alpha[i] lives on lane 3i
    const float a0 = __shfl(alpha, 9 * g + 0);
    const float a1 = __shfl(alpha, 9 * g + 3);
    const float a2 = __shfl(alpha, 9 * g + 6);
    const float x0 = a0 + t0, x1 = a1 + t1, x2 = a2 + t2;
    float mp = fmaxf(x0, fmaxf(x1, x2));
    float sp = __expf(x0 - mp) + __expf(x1 - mp) + __expf(x2 - mp);
    // combine the 3 partials for tag jj across lanes 3jj + {0,1,2}
    const int base = 3 * jj;
    const float mA = __shfl(mp, base + ((g + 1) % 3));
    const float sA = __shfl(sp, base + ((g + 1) % 3));
    const float mB = __shfl(mp, base + ((g + 2) % 3));
    const float sB = __shfl(sp, base + ((g + 2) % 3));
    const float M = fmaxf(mp, fmaxf(mA, mB));
    const float Sm = sp * __expf(mp - M) + sA * __expf(mA - M) + sB * __expf(mB - M);
    const float nxt = ej + M + __logf(Sm);
    alpha = mt ? nxt : alpha;
  }

  // logZ = logsumexp_j(alpha[j] + end[j]); alpha[j] lives on lane 3j
  const float ve = alpha + end_t[jj];
  float m = -INFINITY;
  #pragma unroll
  for (int i = 0; i < 9; ++i) m = fmaxf(m, __shfl(ve, 3 * i));
  float s = 0.0f;
  #pragma unroll
  for (int i = 0; i < 9; ++i) s += __expf(__shfl(ve, 3 * i) - m);
  const float logZ = m + __logf(s);

  // Gold-path score (lanes stride over time, wave reduce)
  float emit_s = 0.0f, tr_s = 0.0f;
  int msum = 0;
  for (int t = lane; t < SS; t += 32) {
    const int mt = mb[t];
    if (mt) {
      const int lt = lb[t];
      emit_s += eb[t * LP + lt];
      msum += 1;
      if (t >= 1) tr_s += trans[lb[t - 1] * LL + lt];
    }
  }
  #pragma unroll
  for (int o = 16; o; o >>= 1) {
    emit_s += __shfl_xor(emit_s, o);
    tr_s   += __shfl_xor(tr_s, o);
    msum   += __shfl_xor(msum, o);
  }
  const int seq_end = msum - 1;
  const float score = start_t[lb[0]] + emit_s + tr_s + end_t[lb[seq_end]];
  if (lane == 0) llh[b] = score - logZ;
}

// ---------------------------------------------------------------------------
// Kernel 3: loss = -mean(llh). Deterministic single-block reduce.
// ---------------------------------------------------------------------------
__global__ __launch_bounds__(64) void reduce_loss(const float* __restrict__ llh,
                                                  float* __restrict__ out) {
  __shared__ float sm[BB];
  sm[threadIdx.x] = llh[threadIdx.x];
  __syncthreads();
  if (threadIdx.x == 0) {
    float s = 0.0f;
    #pragma unroll
    for (int i = 0; i < BB; ++i) s += sm[i];
    out[0] = -s / (float)BB;
  }
}

extern "C" void kernel_launch(void* const* d_in, const int* in_sizes, int n_in,
                              void* d_out, int out_size, void* d_ws, size_t ws_size,
                              hipStream_t stream) {
  (void)in_sizes; (void)n_in; (void)out_size; (void)ws_size;
  const float* hidden = (const float*)d_in[0];
  const float* W      = (const float*)d_in[1];
  const float* bias   = (const float*)d_in[2];
  const float* start_t= (const float*)d_in[3];
  const float* end_t  = (const float*)d_in[4];
  const float* trans  = (const float*)d_in[5];
  const int*   labels = (const int*)d_in[6];
  const int*   mask   = (const int*)d_in[7];

  float* ews = (float*)d_ws;                                          // 2 MB
  float* llh = (float*)((char*)d_ws + (size_t)BB * SS * LP * sizeof(float));

  // 32768 rows / (4 waves * 16 rows) = 512 blocks
  emis_gemm_wmma<<<512, 128, 0, stream>>>(hidden, W, bias, ews);
  crf_scan<<<BB, 32, 0, stream>>>(ews, start_t, end_t, trans, labels, mask, llh);
  reduce_loss<<<1, BB, 0, stream>>>(llh, (float*)d_out);
}